// HeteroGCN_84988812853629
// MI455X (gfx1250) — compile-verified
//
#include <hip/hip_runtime.h>

typedef __attribute__((ext_vector_type(2))) float v2f;
typedef __attribute__((ext_vector_type(8))) float v8f;

#define N_REL 3
#define IN_F  256
#define HID_F 128
#define N_CLS 2
#define LDSK  260   // LDS row stride in floats: 260 mod 64 == 4 -> conflict-free b64 reads

__device__ __forceinline__ void atomAddF(float* p, float v) {
    __hip_atomic_fetch_add(p, v, __ATOMIC_RELAXED, __HIP_MEMORY_SCOPE_AGENT);
}

// ---------------- utility kernels ----------------

__global__ void zero_k(float* __restrict__ p, int n) {
    int i = blockIdx.x * blockDim.x + threadIdx.x;
    if (i < n) p[i] = 0.0f;
}

// degrees as float histograms (exact for counts < 2^24)
__global__ void deg_k(const long long* __restrict__ edges,
                      float* __restrict__ rinv_out, float* __restrict__ rinv_in,
                      int N, int E) {
    int idx = blockIdx.x * blockDim.x + threadIdx.x;
    if (idx >= N_REL * E) return;
    int r = idx / E;
    int e = idx - r * E;
    int s = (int)edges[((size_t)r * 2 + 0) * (size_t)E + e];
    int d = (int)edges[((size_t)r * 2 + 1) * (size_t)E + e];
    atomAddF(rinv_out + (size_t)r * N + s, 1.0f);
    atomAddF(rinv_in  + (size_t)r * N + d, 1.0f);
}

__global__ void rinv_k(float* __restrict__ p, int n) {
    int i = blockIdx.x * blockDim.x + threadIdx.x;
    if (i < n) p[i] = rsqrtf(fmaxf(p[i], 1.0f));
}

// W1 (r,k,n) -> Wt (r,n,k) so B fragments are contiguous 8B per lane
__global__ void tw_k(const float* __restrict__ W1, float* __restrict__ Wt) {
    int i = blockIdx.x * blockDim.x + threadIdx.x;
    if (i >= N_REL * IN_F * HID_F) return;
    int r = i / (IN_F * HID_F);
    int rem = i - r * (IN_F * HID_F);
    int k = rem / HID_F;
    int n = rem - k * HID_F;
    Wt[(size_t)r * IN_F * HID_F + (size_t)n * IN_F + k] = W1[i];
}

// h1 pre-loaded with sum_r b1[r]  (bias added once per node)
__global__ void inith1_k(const float* __restrict__ b1, float* __restrict__ h1, int n) {
    int i = blockIdx.x * blockDim.x + threadIdx.x;
    if (i >= n) return;
    int c = i & (HID_F - 1);
    h1[i] = b1[c] + b1[HID_F + c] + b1[2 * HID_F + c];
}

__global__ void initout_k(const float* __restrict__ b2, float* __restrict__ out, int n) {
    int i = blockIdx.x * blockDim.x + threadIdx.x;
    if (i >= n) return;
    int c = i & (N_CLS - 1);
    out[i] = b2[c] + b2[N_CLS + c] + b2[2 * N_CLS + c];
}

// ---------------- layer 1: fp32 WMMA GEMM  feat = (x * rinv_out) @ W1r ----------------
// block = 256 threads = 8 waves, computes 16 rows x 128 cols; wave w -> cols [16w,16w+16)
__global__ __launch_bounds__(256) void gemm1_k(const float* __restrict__ x,
                                               const float* __restrict__ Wt,   // [128][256]
                                               const float* __restrict__ rinv,
                                               float* __restrict__ feat) {
    __shared__ float xs[16 * LDSK];
    const int t = threadIdx.x;
    const int row0 = blockIdx.x * 16;

    // stage 16x256 x-tile scaled by rsqrt(deg_out): coalesced 16-float chunks
    {
        int row = t >> 4;        // 0..15
        int chunk = t & 15;      // 0..15, 16 floats each
        float sc = rinv[row0 + row];
        const float4* gs = (const float4*)(x + (size_t)(row0 + row) * IN_F + chunk * 16);
        float4* ls = (float4*)(xs + row * LDSK + chunk * 16);
        #pragma unroll
        for (int j = 0; j < 4; ++j) {
            float4 v = gs[j];
            v.x *= sc; v.y *= sc; v.z *= sc; v.w *= sc;
            ls[j] = v;
        }
    }
    __syncthreads();

    const int lane = t & 31;
    const int w = t >> 5;
    const int ln = lane & 15;
    const int hi = lane >> 4;
    const int n0 = w * 16;

    const float* ap = xs + ln * LDSK + 2 * hi;                       // A: row ln, K pair
    const float* bp = Wt + (size_t)(n0 + ln) * IN_F + 2 * hi;        // B: col n0+ln, K pair

    v8f acc = {};
    #pragma unroll 8
    for (int k0 = 0; k0 < IN_F; k0 += 4) {
        v2f a = *(const v2f*)(ap + k0);
        v2f b = *(const v2f*)(bp + k0);
        acc = __builtin_amdgcn_wmma_f32_16x16x4_f32(false, a, false, b,
                                                    (short)0, acc, false, false);
    }

    // C/D layout: VGPR v, lanes 0-15 -> M=v, lanes 16-31 -> M=v+8
    float* op = feat + (size_t)(row0 + 8 * hi) * HID_F + n0 + ln;
    #pragma unroll
    for (int v = 0; v < 8; ++v) op[(size_t)v * HID_F] = acc[v];
}

// ---------------- layer 1 scatter: one wave per edge, 4 floats/lane ----------------
__global__ __launch_bounds__(256) void scatter1_k(const float* __restrict__ feat,
                                                  const long long* __restrict__ src,
                                                  const long long* __restrict__ dst,
                                                  const float* __restrict__ rinv_in,
                                                  float* __restrict__ h1, int E) {
    int wid = (int)((blockIdx.x * 256u + threadIdx.x) >> 5);
    if (wid >= E) return;
    int lane = threadIdx.x & 31;
    int s = (int)src[wid];
    int d = (int)dst[wid];
    float sc = rinv_in[d];
    float4 f = *(const float4*)(feat + (size_t)s * HID_F + lane * 4);
    float* o = h1 + (size_t)d * HID_F + lane * 4;
    atomAddF(o + 0, f.x * sc);
    atomAddF(o + 1, f.y * sc);
    atomAddF(o + 2, f.z * sc);
    atomAddF(o + 3, f.w * sc);
}

// ---------------- layer 2 GEMM (K=128, N=2): wave per node, shuffle reduce ----------------
__global__ __launch_bounds__(256) void gemm2_k(const float* __restrict__ h1,
                                               const float* __restrict__ W2r,  // [128][2]
                                               const float* __restrict__ rinv,
                                               float* __restrict__ feat2, int N) {
    int wid = (int)((blockIdx.x * 256u + threadIdx.x) >> 5);
    if (wid >= N) return;
    int lane = threadIdx.x & 31;
    float4 hv = *(const float4*)(h1 + (size_t)wid * HID_F + lane * 4);
    float v0 = fmaxf(hv.x, 0.0f), v1 = fmaxf(hv.y, 0.0f);
    float v2 = fmaxf(hv.z, 0.0f), v3 = fmaxf(hv.w, 0.0f);
    float2 w0 = *(const float2*)(W2r + (lane * 4 + 0) * 2);
    float2 w1 = *(const float2*)(W2r + (lane * 4 + 1) * 2);
    float2 w2 = *(const float2*)(W2r + (lane * 4 + 2) * 2);
    float2 w3 = *(const float2*)(W2r + (lane * 4 + 3) * 2);
    float a0 = v0 * w0.x + v1 * w1.x + v2 * w2.x + v3 * w3.x;
    float a1 = v0 * w0.y + v1 * w1.y + v2 * w2.y + v3 * w3.y;
    #pragma unroll
    for (int o = 16; o > 0; o >>= 1) {
        a0 += __shfl_down(a0, o, 32);
        a1 += __shfl_down(a1, o, 32);
    }
    if (lane == 0) {
        float sc = rinv[wid];
        feat2[(size_t)wid * 2 + 0] = a0 * sc;
        feat2[(size_t)wid * 2 + 1] = a1 * sc;
    }
}

// ---------------- layer 2 scatter: thread per edge ----------------
__global__ void scatter2_k(const float* __restrict__ feat2,
                           const long long* __restrict__ src,
                           const long long* __restrict__ dst,
                           const float* __restrict__ rinv_in,
                           float* __restrict__ out, int E) {
    int e = blockIdx.x * blockDim.x + threadIdx.x;
    if (e >= E) return;
    int s = (int)src[e];
    int d = (int)dst[e];
    float sc = rinv_in[d];
    float2 f = *(const float2*)(feat2 + (size_t)s * 2);
    atomAddF(out + (size_t)d * 2 + 0, f.x * sc);
    atomAddF(out + (size_t)d * 2 + 1, f.y * sc);
}

// ---------------- host launch ----------------
extern "C" void kernel_launch(void* const* d_in, const int* in_sizes, int n_in,
                              void* d_out, int out_size, void* d_ws, size_t ws_size,
                              hipStream_t stream) {
    (void)n_in; (void)out_size; (void)ws_size;
    const float*     x     = (const float*)d_in[0];
    const long long* edges = (const long long*)d_in[1];
    const float*     W1    = (const float*)d_in[2];
    const float*     b1    = (const float*)d_in[3];
    const float*     W2    = (const float*)d_in[4];
    const float*     b2    = (const float*)d_in[5];
    float* out = (float*)d_out;

    const int N = in_sizes[0] / IN_F;           // 100000
    const int E = in_sizes[1] / (N_REL * 2);    // 1600000

    float* ws       = (float*)d_ws;
    float* rinv_out = ws;                                   // [3][N]
    float* rinv_in  = rinv_out + (size_t)N_REL * N;         // [3][N]
    float* Wt       = rinv_in  + (size_t)N_REL * N;         // [3][128][256]
    float* feat     = Wt   + (size_t)N_REL * IN_F * HID_F;  // [N][128]
    float* h1       = feat + (size_t)N * HID_F;             // [N][128]
    float* feat2    = h1   + (size_t)N * HID_F;             // [N][2]

    const int nDeg = 2 * N_REL * N;
    zero_k<<<(nDeg + 255) / 256, 256, 0, stream>>>(rinv_out, nDeg);
    deg_k<<<(N_REL * E + 255) / 256, 256, 0, stream>>>(edges, rinv_out, rinv_in, N, E);
    rinv_k<<<(nDeg + 255) / 256, 256, 0, stream>>>(rinv_out, nDeg);
    tw_k<<<(N_REL * IN_F * HID_F + 255) / 256, 256, 0, stream>>>(W1, Wt);

    const int nH1 = N * HID_F;
    inith1_k<<<(nH1 + 255) / 256, 256, 0, stream>>>(b1, h1, nH1);

    for (int r = 0; r < N_REL; ++r) {
        const long long* srcr = edges + ((size_t)r * 2 + 0) * (size_t)E;
        const long long* dstr = edges + ((size_t)r * 2 + 1) * (size_t)E;
        gemm1_k<<<N / 16, 256, 0, stream>>>(x, Wt + (size_t)r * IN_F * HID_F,
                                            rinv_out + (size_t)r * N, feat);
        scatter1_k<<<(E * 8 + 7) / 8, 256, 0, stream>>>(feat, srcr, dstr,
                                                        rinv_in + (size_t)r * N, h1, E);
    }

    initout_k<<<(N * N_CLS + 255) / 256, 256, 0, stream>>>(b2, out, N * N_CLS);

    for (int r = 0; r < N_REL; ++r) {
        const long long* srcr = edges + ((size_t)r * 2 + 0) * (size_t)E;
        const long long* dstr = edges + ((size_t)r * 2 + 1) * (size_t)E;
        gemm2_k<<<(N + 7) / 8, 256, 0, stream>>>(h1, W2 + (size_t)r * HID_F * N_CLS,
                                                 rinv_out + (size_t)r * N, feat2, N);
        scatter2_k<<<(E + 255) / 256, 256, 0, stream>>>(feat2, srcr, dstr,
                                                        rinv_in + (size_t)r * N, out, E);
    }
}